// MSMDDeformRegionAttn_19791209300007
// MI455X (gfx1250) — compile-verified
//
#include <hip/hip_runtime.h>
#include <hip/hip_bf16.h>
#include <math.h>

#define NB      4
#define LQ      300
#define CDIM    256
#define NHEADS  8
#define NLEV    4
#define NPTS    4
#define ROIW    7
#define DH      32
#define LEN_IN  19947
#define ROI_DIM (ROIW * ROIW * CDIM)   /* 12544 */
#define NPDIM   (NHEADS * NLEV * NPTS) /* 128   */

#define KC  64   /* K-chunk staged in LDS (all Ks are multiples of 64)   */
#define NT  64   /* N-tile per wave (4x 16x16 accumulators)              */
#define SBP 80   /* padded LDS row pitch (floats): rows k and k+2 land   */
                 /* 32 banks apart -> conflict-free dual-row B reads     */

typedef float v2f __attribute__((ext_vector_type(2)));
typedef float v8f __attribute__((ext_vector_type(8)));
typedef unsigned int u32x4 __attribute__((ext_vector_type(4)));
typedef int i32x8 __attribute__((ext_vector_type(8)));
typedef int i32x4 __attribute__((ext_vector_type(4)));

#if defined(__has_builtin)
#if __has_builtin(__builtin_amdgcn_tensor_load_to_lds)
#define HAVE_TDM 1
#endif
#endif
#ifndef HAVE_TDM
#define HAVE_TDM 0
#endif

// ---------------------------------------------------------------------------
// Zero-padded bilinear sample matching grid_sample(align_corners=False,
// padding_mode='zeros'): clip indices for the read, multiply weight by valid.
// ---------------------------------------------------------------------------
__device__ __forceinline__ float bilinear_zero_pad(const float* __restrict__ p,
                                                   int H, int W, int chanStride,
                                                   float y, float x) {
  float y0f = floorf(y), x0f = floorf(x);
  int y0 = (int)y0f, x0 = (int)x0f;
  float fy = y - y0f, fx = x - x0f;
  float r = 0.f;
#pragma unroll
  for (int dy = 0; dy < 2; ++dy) {
    int yi = y0 + dy;
    float wy = dy ? fy : 1.f - fy;
    bool vy = (yi >= 0) && (yi < H);
    int yc = yi < 0 ? 0 : (yi > H - 1 ? H - 1 : yi);
#pragma unroll
    for (int dx = 0; dx < 2; ++dx) {
      int xi = x0 + dx;
      float wx = dx ? fx : 1.f - fx;
      bool vx = (xi >= 0) && (xi < W);
      int xc = xi < 0 ? 0 : (xi > W - 1 ? W - 1 : xi);
      float w = (vy && vx) ? (wy * wx) : 0.f;
      r += w * p[((size_t)yc * W + xc) * chanStride];
    }
  }
  return r;
}

// ---------------------------------------------------------------------------
// fp32 WMMA GEMM: C[M,N] = act(A[M,K] @ B[K,N] + bias[N]) with optional row
// mask. Block = 8 waves -> 128x64 block tile; each wave owns a 16x64 strip
// (4x v_wmma_f32_16x16x4_f32 accumulators). B K-chunks (KCxNT) are staged in
// LDS once per block via the Tensor Data Mover (TDM descriptor with LDS
// padding -> SBP=80 pitch), wave 0 issues tensor_load_to_lds and waits on
// TENSORcnt. A fragments stream from global (L2-resident) as aligned float2
// per the ISA fp32 16x4 A layout: lanes 0-15 hold K=k0,k0+1; lanes 16-31
// hold K=k0+2,k0+3. Requires K % KC == 0, N % NT == 0.
// ---------------------------------------------------------------------------
__global__ __launch_bounds__(256) void gemm_wmma_f32(
    const float* __restrict__ A, const float* __restrict__ B,
    const float* __restrict__ bias, float* __restrict__ C,
    int M, int N, int K, int act, const unsigned char* __restrict__ mask) {
  __shared__ float s_b[KC * SBP];

  const int lane  = threadIdx.x & 31;
  const int wave  = threadIdx.x >> 5;
  const int mtile = blockIdx.x * 128 + wave * 16;
  const int ntile = blockIdx.y * NT;

  const int r16 = lane & 15;
  const int kb  = (lane >> 4) << 1;  // 0 for lanes 0-15, 2 for lanes 16-31
  int mrow = mtile + r16;
  if (mrow > M - 1) mrow = M - 1;    // clamp tail rows; stores are guarded
  const float* arow = A + (size_t)mrow * K + kb;

  v8f acc[4] = {v8f{}, v8f{}, v8f{}, v8f{}};

  for (int kc = 0; kc < K; kc += KC) {
#if HAVE_TDM
    if (wave == 0) {
      // --- TDM: DMA B[kc..kc+KC)[ntile..ntile+NT) -> s_b with row padding ---
      unsigned long long ga =
          (unsigned long long)(const void*)(B + (size_t)kc * N + ntile);
      unsigned lds_off = (unsigned)(size_t)(void*)s_b;
      u32x4 g0;
      g0[0] = 1u;                                   // count=1, user descriptor
      g0[1] = lds_off;                              // lds_addr (bytes)
      g0[2] = (unsigned)(ga & 0xffffffffu);         // global_addr[31:0]
      g0[3] = ((unsigned)(ga >> 32) & 0x01ffffffu)  // global_addr[56:32]
              | 0x80000000u;                        // type=2 ("image")
      i32x8 g1;
      g1[0] = (2 << 16)        // data_size = 4B
            | (1 << 20)        // pad_enable
            | (5 << 22)        // pad_interval: 64 DWORDs (one tile row)
            | (15 << 25);      // pad_amount: 16 DWORDs -> SBP=80 pitch
      g1[1] = (int)(((unsigned)N & 0xffffu) << 16);          // tensor_dim0 lo
      g1[2] = (int)(((unsigned)N >> 16) |
                    (((unsigned)K & 0xffffu) << 16));        // d0 hi | d1 lo
      g1[3] = (int)(((unsigned)K >> 16) | (NT << 16));       // d1 hi | tile_d0
      g1[4] = KC;                                            // tile_dim1
      g1[5] = N;                                             // dim0_stride lo
      g1[6] = 0;
      g1[7] = 0;
      i32x4 z4 = {0, 0, 0, 0};
      i32x8 z8 = {0, 0, 0, 0, 0, 0, 0, 0};
      __builtin_amdgcn_tensor_load_to_lds(g0, g1, z4, z4, z8, 0);
      __builtin_amdgcn_s_wait_tensorcnt(0);
    }
#else
    for (int i = threadIdx.x; i < KC * NT; i += 256)
      s_b[(i >> 6) * SBP + (i & 63)] =
          B[(size_t)(kc + (i >> 6)) * N + ntile + (i & 63)];
#endif
    __syncthreads();
    __builtin_prefetch(arow + kc + KC, 0, 0);  // global_prefetch next A chunk

#pragma unroll 4
    for (int k0 = 0; k0 < KC; k0 += 4) {
      v2f a = *(const v2f*)(arow + kc + k0);  // K=k0+kb, k0+kb+1 (8B aligned)
      const float* br0 = &s_b[(k0 + kb) * SBP + r16];
      const float* br1 = br0 + SBP;
      v2f b0, b1, b2, b3;
      b0.x = br0[0];  b0.y = br1[0];
      b1.x = br0[16]; b1.y = br1[16];
      b2.x = br0[32]; b2.y = br1[32];
      b3.x = br0[48]; b3.y = br1[48];
      acc[0] = __builtin_amdgcn_wmma_f32_16x16x4_f32(false, a, false, b0,
                                                     (short)0, acc[0], false, false);
      acc[1] = __builtin_amdgcn_wmma_f32_16x16x4_f32(false, a, false, b1,
                                                     (short)0, acc[1], false, false);
      acc[2] = __builtin_amdgcn_wmma_f32_16x16x4_f32(false, a, false, b2,
                                                     (short)0, acc[2], false, false);
      acc[3] = __builtin_amdgcn_wmma_f32_16x16x4_f32(false, a, false, b3,
                                                     (short)0, acc[3], false, false);
    }
    __syncthreads();  // protect s_b before next chunk's DMA overwrites it
  }

  const int mbase = mtile + ((lane >> 4) << 3);  // +8 for upper half lanes
#pragma unroll
  for (int j = 0; j < 4; ++j) {
    const int nc = ntile + j * 16 + r16;
    const float bn = bias[nc];
#pragma unroll
    for (int r = 0; r < 8; ++r) {
      int m = mbase + r;
      if (m < M) {
        float v = acc[j][r] + bn;
        if (act == 1) v = tanhf(v);
        if (mask && mask[m]) v = 0.f;
        C[(size_t)m * N + nc] = v;
      }
    }
  }
}

// ---------------------------------------------------------------------------
// ROI align on level-0 feature map: one block per (n, q, sy); thread = channel.
// ---------------------------------------------------------------------------
__global__ __launch_bounds__(256) void roi_align_kernel(
    const float* __restrict__ value, const float* __restrict__ refpts,
    const int* __restrict__ shapes, float* __restrict__ roi) {
  int b  = blockIdx.x;
  int sy = b % ROIW;
  int q  = (b / ROIW) % LQ;
  int n  = b / (ROIW * LQ);
  int c  = threadIdx.x;

  int H0 = shapes[0], W0 = shapes[1];
  const float* rp = refpts + ((size_t)(n * LQ + q) * NLEV + 0) * 4;
  float cx = rp[0], cy = rp[1], w = rp[2], h = rp[3];
  float x1  = (cx - 0.5f * w) * W0 - 0.5f;
  float y1  = (cy - 0.5f * h) * H0 - 0.5f;
  float bw_ = (w * W0) * (1.f / ROIW);
  float bh_ = (h * H0) * (1.f / ROIW);
  float yy  = y1 + bh_ * (sy + 0.5f);

  const float* base = value + (size_t)n * LEN_IN * CDIM + c;  // level-0 start=0
  float* out = roi + ((size_t)(n * LQ + q) * (ROIW * ROIW) + sy * ROIW) * CDIM + c;

  for (int sx = 0; sx < ROIW; ++sx) {
    float xx = x1 + bw_ * (sx + 0.5f);
    out[(size_t)sx * CDIM] = bilinear_zero_pad(base, H0, W0, CDIM, yy, xx);
  }
}

// ---------------------------------------------------------------------------
// Deformable sampling: one block per (n,q). Per-head softmax over 16 (L*P),
// points = ref_center + tanh_offsets * ref_wh * 0.5, bilinear per (h,d).
// pr input already tanh'ed by the GEMM epilogue (act=1).
// ---------------------------------------------------------------------------
__global__ __launch_bounds__(256) void sample_kernel(
    const float* __restrict__ value, const float* __restrict__ refpts,
    const int* __restrict__ shapes, const int* __restrict__ starts,
    const float* __restrict__ pr, const float* __restrict__ logits,
    float* __restrict__ out_mid) {
  __shared__ float s_pr[NPDIM * 2];
  __shared__ float s_attn[NPDIM];
  __shared__ float s_ref[NLEV * 4];

  int nq  = blockIdx.x;
  int tid = threadIdx.x;
  s_pr[tid] = pr[(size_t)nq * (NPDIM * 2) + tid];
  if (tid < NPDIM)    s_attn[tid] = logits[(size_t)nq * NPDIM + tid];
  if (tid < NLEV * 4) s_ref[tid]  = refpts[(size_t)nq * NLEV * 4 + tid];
  __syncthreads();

  if (tid < NHEADS) {  // per-head softmax over 16 logits (serial, trivial)
    float mx = -1e30f;
    for (int j = 0; j < NLEV * NPTS; ++j) mx = fmaxf(mx, s_attn[tid * 16 + j]);
    float e[NLEV * NPTS];
    float sum = 0.f;
    for (int j = 0; j < NLEV * NPTS; ++j) {
      e[j] = __expf(s_attn[tid * 16 + j] - mx);
      sum += e[j];
    }
    float inv = 1.f / sum;
    for (int j = 0; j < NLEV * NPTS; ++j) s_attn[tid * 16 + j] = e[j] * inv;
  }
  __syncthreads();

  int h = tid >> 5;   // head
  int d = tid & 31;   // channel within head (Dh=32) -> coalesced value loads
  int n = nq / LQ;
  float acc = 0.f;
#pragma unroll
  for (int l = 0; l < NLEV; ++l) {
    int Hl = shapes[2 * l], Wl = shapes[2 * l + 1];
    int st = starts[l];
    const float* base = value + ((size_t)n * LEN_IN + st) * CDIM + h * DH + d;
    float rcx = s_ref[l * 4 + 0], rcy = s_ref[l * 4 + 1];
    float rw  = s_ref[l * 4 + 2], rh  = s_ref[l * 4 + 3];
#pragma unroll
    for (int p = 0; p < NPTS; ++p) {
      int idx  = (h * NLEV + l) * NPTS + p;
      float ox = s_pr[idx * 2 + 0], oy = s_pr[idx * 2 + 1];
      float px = (rcx + ox * rw * 0.5f) * Wl - 0.5f;
      float py = (rcy + oy * rh * 0.5f) * Hl - 0.5f;
      acc += s_attn[h * 16 + l * NPTS + p] *
             bilinear_zero_pad(base, Hl, Wl, CDIM, py, px);
    }
  }
  out_mid[(size_t)nq * CDIM + h * DH + d] = acc;
}

// ---------------------------------------------------------------------------
// Launch: 6-stage pipeline on `stream`. Workspace (float):
// value | roi_flat | pr_lin | logits | out_mid  (~145 MB, L2-resident).
// ---------------------------------------------------------------------------
extern "C" void kernel_launch(void* const* d_in, const int* in_sizes, int n_in,
                              void* d_out, int out_size, void* d_ws, size_t ws_size,
                              hipStream_t stream) {
  (void)in_sizes; (void)n_in; (void)out_size; (void)ws_size;

  const float*         refpts = (const float*)d_in[1];
  const float*         inflat = (const float*)d_in[2];
  const int*           shapes = (const int*)d_in[3];
  const int*           starts = (const int*)d_in[4];
  const unsigned char* pmask  = (const unsigned char*)d_in[5];
  const float* Wv = (const float*)d_in[6];
  const float* bv = (const float*)d_in[7];
  const float* Wp = (const float*)d_in[8];
  const float* bp = (const float*)d_in[9];
  const float* Ww = (const float*)d_in[10];
  const float* bw = (const float*)d_in[11];
  const float* Wo = (const float*)d_in[12];
  const float* bo = (const float*)d_in[13];
  float* out = (float*)d_out;

  float* ws      = (float*)d_ws;
  float* value   = ws;                                        // 79788*256
  float* roi     = value   + (size_t)NB * LEN_IN * CDIM;      // 1200*12544
  float* pr_lin  = roi     + (size_t)NB * LQ * ROI_DIM;       // 1200*256
  float* logits  = pr_lin  + (size_t)NB * LQ * (NPDIM * 2);   // 1200*128
  float* out_mid = logits  + (size_t)NB * LQ * NPDIM;         // 1200*256

  const int M1 = NB * LEN_IN;  // 79788 rows of value projection
  const int MQ = NB * LQ;      // 1200 query rows
  dim3 blk(256);

  // 1) value = mask(input_flatten @ Wv + bv)
  gemm_wmma_f32<<<dim3((M1 + 127) / 128, CDIM / NT), blk, 0, stream>>>(
      inflat, Wv, bv, value, M1, CDIM, CDIM, /*act=*/0, pmask);

  // 2) 7x7 ROI align on level-0 of value
  roi_align_kernel<<<NB * LQ * ROIW, blk, 0, stream>>>(value, refpts, shapes, roi);

  // 3) pr = tanh(roi_flat @ Wp + bp)
  gemm_wmma_f32<<<dim3((MQ + 127) / 128, (NPDIM * 2) / NT), blk, 0, stream>>>(
      roi, Wp, bp, pr_lin, MQ, NPDIM * 2, ROI_DIM, /*act=*/1, nullptr);

  // 4) logits = roi_flat @ Ww + bw
  gemm_wmma_f32<<<dim3((MQ + 127) / 128, NPDIM / NT), blk, 0, stream>>>(
      roi, Ww, bw, logits, MQ, NPDIM, ROI_DIM, /*act=*/0, nullptr);

  // 5) softmax + deformable multi-scale bilinear sampling
  sample_kernel<<<NB * LQ, blk, 0, stream>>>(value, refpts, shapes, starts,
                                             pr_lin, logits, out_mid);

  // 6) out = out_mid @ Wo + bo
  gemm_wmma_f32<<<dim3((MQ + 127) / 128, CDIM / NT), blk, 0, stream>>>(
      out_mid, Wo, bo, out, MQ, CDIM, CDIM, /*act=*/0, nullptr);
}